// Fastsum_40699110097513
// MI455X (gfx1250) — compile-verified
//
#include <hip/hip_runtime.h>
#include <math.h>

#define DIM   64
#define NPTS  8192
#define NSL   32
#define KH    512                     // half spectrum, k = 0..511 (pair-symmetric)
#define NCH   8                       // point-chunks for adjoint stage (1024 pts each)
#define NSG   8                       // slice-groups for synthesis stage (4 slices each)
#define LOG_CONSTF (-41.460867206134486f)   // 32*ln(pi) - lgamma(32)
#define TWO_PI_F   6.283185307179586f
#define PI_SQ_F    9.869604401089358f

// ---- workspace layout (float offsets) ----
#define BMAX_OFF 0                          // 64 block maxima
#define SF_OFF   64                         // 2*pi*sf
#define KFT2_OFF 128                        // kft2[512]
#define THX_OFF  640                        // thetaX[32][8192] slice-major
#define THY_OFF  (THX_OFF + NSL*NPTS)       // thetaY[32][8192]
#define PRE_OFF  (THY_OFF + NSL*NPTS)       // Re partials [32][NCH][512]
#define PIM_OFF  (PRE_OFF + NSL*NCH*KH)     // Im partials [32][NCH][512]
#define U_OFF    (PIM_OFF + NSL*NCH*KH)     // u[32][512] = kft2*Re(a)
#define V_OFF    (U_OFF + NSL*KH)           // v[32][512] = kft2*Im(a)
#define SOUT_OFF (V_OFF + NSL*KH)           // synth partials [NSG][8192]

typedef __attribute__((ext_vector_type(2))) float v2f;
typedef __attribute__((ext_vector_type(4))) float v4f;
typedef __attribute__((ext_vector_type(8))) float v8f;

// ---------------- K1: max squared norm over x and y ----------------
__global__ void k_maxnorm(const float* __restrict__ x, const float* __restrict__ y,
                          float* __restrict__ ws) {
    int idx = blockIdx.x * blockDim.x + threadIdx.x;           // 0..16383
    const float* p = (idx < NPTS) ? (x + (size_t)idx * DIM)
                                  : (y + (size_t)(idx - NPTS) * DIM);
    float s = 0.f;
    #pragma unroll 8
    for (int d = 0; d < DIM; ++d) { float v = p[d]; s = fmaf(v, v, s); }
    __shared__ float red[256];
    red[threadIdx.x] = s;
    __syncthreads();
    for (int off = 128; off > 0; off >>= 1) {
        if (threadIdx.x < off)
            red[threadIdx.x] = fmaxf(red[threadIdx.x], red[threadIdx.x + off]);
        __syncthreads();
    }
    if (threadIdx.x == 0) ws[BMAX_OFF + blockIdx.x] = red[0];
}

// ---------------- K2: sf + kernel Fourier coefficients (log-space) ----------------
__global__ void k_spectrum(const float* __restrict__ scale_p, float* __restrict__ ws) {
    __shared__ float sm[66];
    int t = threadIdx.x;                                        // 512 threads
    if (t < 64) sm[t] = ws[BMAX_OFF + t];
    __syncthreads();
    if (t == 0) {
        float m = sm[0];
        for (int i = 1; i < 64; ++i) m = fmaxf(m, sm[i]);
        float sf = 0.3f / sqrtf(m);
        float s  = scale_p[0] * sf;                             // rescaled bandwidth
        sm[64]   = s * s;
        ws[SF_OFF] = TWO_PI_F * sf;                             // projection scale
    }
    __syncthreads();
    float s2 = sm[64];
    float val = 0.f;
    if (t >= 1) {
        float kf   = (float)t;
        float logv = LOG_CONSTF + 32.f * logf(TWO_PI_F * s2)
                   + 63.f * logf(kf) - 2.f * PI_SQ_F * s2 * kf * kf;
        val = 2.f * expf(logv);                                 // factor 2: +-k pairing
    }
    ws[KFT2_OFF + t] = val;
}

// ---------------- K3: projections theta = 2*pi*sf*(points @ xis^T) via WMMA f32 ----
// One wave -> one 16(points) x 16(slices) f32 tile, K=64 in 16 chunks of 4.
__global__ void k_project(const float* __restrict__ x, const float* __restrict__ y,
                          const float* __restrict__ xis, float* __restrict__ ws) {
    const float tps = ws[SF_OFF];                               // 2*pi*sf (uniform)
    int lane = threadIdx.x & 31;
    int wave = threadIdx.x >> 5;
    int g    = blockIdx.x * 8 + wave;                           // 0..2047
    int mat  = g >> 10;                                         // 0 = x, 1 = y
    int tile = g & 1023;
    int row0 = (tile >> 1) * 16;                                // point tile
    int col0 = (tile & 1) * 16;                                 // slice tile
    const float* src = mat ? y : x;
    float* dst = ws + (mat ? THY_OFF : THX_OFF);

    int mrow = lane & 15;                                       // M (A) / N (B) index
    int dhi  = (lane >> 4) << 1;                                // K sub-pair: 0 or 2
    v8f c = {};
    #pragma unroll
    for (int kk = 0; kk < 16; ++kk) {
        int d0 = kk * 4 + dhi;
        // A 16x4 f32: VGPR0 = K even, VGPR1 = K odd; lane halves split K pairs
        const float* ap = src + (size_t)(row0 + mrow) * DIM + d0;
        v2f a = { ap[0], ap[1] };
        // B 4x16 f32: row K striped across lanes; same half-split over K
        const float* bp = xis + (size_t)(col0 + mrow) * DIM + d0;
        v2f b = { bp[0], bp[1] };
        c = __builtin_amdgcn_wmma_f32_16x16x4_f32(false, a, false, b,
                                                  (short)0, c, false, false);
    }
    // C/D: VGPR r -> M=r (lanes 0-15) or M=r+8 (lanes 16-31), N = lane&15
    int m0 = (lane >> 4) << 3;                                  // 0 or 8
    float* o = dst + (size_t)(col0 + mrow) * NPTS + row0 + m0;
    v4f lo = { tps * c[0], tps * c[1], tps * c[2], tps * c[3] };
    v4f hi = { tps * c[4], tps * c[5], tps * c[6], tps * c[7] };
    *(v4f*)(o)     = lo;
    *(v4f*)(o + 4) = hi;
}

// ---------------- K4a: adjoint NDFT partials, thread-owns-k ----------------
// Block = (slice, chunk of 1024 points): re/im accumulate privately in registers,
// partials written per (slice,chunk) -> fully deterministic, no atomics.
__global__ void k_adjoint_part(const float* __restrict__ w, float* __restrict__ ws) {
    int slice = blockIdx.x / NCH;
    int chunk = blockIdx.x % NCH;
    int k     = threadIdx.x;                                    // 0..511
    int p0    = chunk * (NPTS / NCH);                           // 1024 points
    const float* th = ws + THX_OFF + (size_t)slice * NPTS + p0;
    const float* wp = w + p0;
    __shared__ float thl[512];
    __shared__ float wl[512];
    float kf = (float)k;
    float re = 0.f, im = 0.f;
    for (int base = 0; base < NPTS / NCH; base += 512) {
        __syncthreads();
        thl[k] = th[base + k];
        wl[k]  = wp[base + k];
        __syncthreads();
        #pragma unroll 4
        for (int j = 0; j < 512; ++j) {
            float s, c;
            __sincosf(kf * thl[j], &s, &c);
            float wv = wl[j];
            re = fmaf(wv,  c, re);
            im = fmaf(-wv, s, im);                              // e^{-i k theta}
        }
    }
    ws[PRE_OFF + (size_t)(slice * NCH + chunk) * KH + k] = re;
    ws[PIM_OFF + (size_t)(slice * NCH + chunk) * KH + k] = im;
}

// ---------------- K4b: fold chunk partials in fixed order, apply kft2 ----------
__global__ void k_adjoint_reduce(float* __restrict__ ws) {
    int slice = blockIdx.x;
    int k     = threadIdx.x;                                    // 0..511
    float re = 0.f, im = 0.f;
    #pragma unroll
    for (int c = 0; c < NCH; ++c) {
        re += ws[PRE_OFF + (size_t)(slice * NCH + c) * KH + k];
        im += ws[PIM_OFF + (size_t)(slice * NCH + c) * KH + k];
    }
    float kft2 = ws[KFT2_OFF + k];
    ws[U_OFF + slice * KH + k] = kft2 * re;
    ws[V_OFF + slice * KH + k] = kft2 * im;
}

// ---------------- K5a: synthesis NDFT partials, thread-owns-j -------------------
// Block = (j-block, group of 4 slices); private accumulator -> deterministic.
__global__ void k_synth_part(const float* __restrict__ ws_c, float* __restrict__ ws) {
    int jb  = blockIdx.x & 31;
    int grp = blockIdx.x >> 5;                                  // 0..7
    int j   = jb * blockDim.x + threadIdx.x;                    // 0..8191
    __shared__ float ul[KH];
    __shared__ float vl[KH];
    float acc = 0.f;
    for (int p = grp * (NSL / NSG); p < (grp + 1) * (NSL / NSG); ++p) {
        __syncthreads();
        ul[threadIdx.x]       = ws_c[U_OFF + p * KH + threadIdx.x];
        ul[threadIdx.x + 256] = ws_c[U_OFF + p * KH + threadIdx.x + 256];
        vl[threadIdx.x]       = ws_c[V_OFF + p * KH + threadIdx.x];
        vl[threadIdx.x + 256] = ws_c[V_OFF + p * KH + threadIdx.x + 256];
        __syncthreads();
        float phi = ws_c[THY_OFF + (size_t)p * NPTS + j];
        #pragma unroll 4
        for (int k = 1; k < KH; ++k) {
            float s, c;
            __sincosf((float)k * phi, &s, &c);
            acc = fmaf(ul[k],  c, acc);
            acc = fmaf(-vl[k], s, acc);
        }
    }
    ws[SOUT_OFF + (size_t)grp * NPTS + j] = acc;
}

// ---------------- K5b: fold slice-group partials in fixed order, /P ----------
__global__ void k_synth_reduce(float* __restrict__ out, const float* __restrict__ ws) {
    int j = blockIdx.x * blockDim.x + threadIdx.x;              // 0..8191
    float acc = 0.f;
    #pragma unroll
    for (int g = 0; g < NSG; ++g)
        acc += ws[SOUT_OFF + (size_t)g * NPTS + j];
    out[j] = acc * (1.0f / NSL);
}

extern "C" void kernel_launch(void* const* d_in, const int* in_sizes, int n_in,
                              void* d_out, int out_size, void* d_ws, size_t ws_size,
                              hipStream_t stream) {
    const float* x     = (const float*)d_in[0];   // [8192,64]
    const float* y     = (const float*)d_in[1];   // [8192,64]
    const float* wts   = (const float*)d_in[2];   // [8192]
    const float* scale = (const float*)d_in[3];   // [1]
    const float* xis   = (const float*)d_in[4];   // [32,64]
    float* ws  = (float*)d_ws;
    float* out = (float*)d_out;

    k_maxnorm       <<<64,        256, 0, stream>>>(x, y, ws);
    k_spectrum      <<<1,         512, 0, stream>>>(scale, ws);
    k_project       <<<256,       256, 0, stream>>>(x, y, xis, ws);  // WMMA f32 16x16x4
    k_adjoint_part  <<<NSL * NCH, 512, 0, stream>>>(wts, ws);        // 4K waves
    k_adjoint_reduce<<<NSL,       512, 0, stream>>>(ws);
    k_synth_part    <<<32 * NSG,  256, 0, stream>>>(ws, ws);         // 2K waves
    k_synth_reduce  <<<32,        256, 0, stream>>>(out, ws);
}